// SparseMSDeformableAttention_84782654423161
// MI455X (gfx1250) — compile-verified
//
#include <hip/hip_runtime.h>
#include <hip/hip_bf16.h>

// ---------------------------------------------------------------------------
// Sparse multi-scale deformable attention for MI455X (gfx1250, wave32).
// All four dense GEMMs run on V_WMMA_F32_16X16X4_F32 (full fp32 matrix pipe,
// matches the fp32 reference numerics). Each wave owns TWO 16x16 C tiles
// (two M-tiles, one N-tile) so every global B load feeds two WMMAs.
// ---------------------------------------------------------------------------

typedef float v2f __attribute__((ext_vector_type(2)));
typedef float v8f __attribute__((ext_vector_type(8)));

#define EMBED_DIM 256
#define N_HEAD 8
#define N_LEVELS 4
#define N_POINTS 4
#define HEAD_DIM 32
#define LEN_Q 13294
#define BS 2
#define KDIM 256  // all GEMMs have K = EMBED_DIM

// ---------------------------------------------------------------------------
// C[M,N] = A[M,K=256] * W[N,K]^T + bias[N], optionally scaled by zeta[n&31].
// Block: 256 threads = 8 waves. Block tile: 32 rows x 128 cols.
// Each wave: one 16-col N-tile, two 16-row M-tiles (two accumulators).
// gridDim.x = N/128, gridDim.y = ceil(M/32).
//
// f32 WMMA operand layout (CDNA5 ISA 7.12.2):
//   A (16x4):  lane l -> M = l&15, holds K = {kb, kb+1}, kb = (l>>4)*2
//   B (4x16):  lane l -> N = l&15, holds K = {kb, kb+1}
//   C/D 16x16: lane l -> N = l&15, VGPR r -> M = r + 8*(l>>4)
// ---------------------------------------------------------------------------
__global__ __launch_bounds__(256)
void wmma_gemm_bias(const float* __restrict__ A,     // [M, 256] row-major
                    const float* __restrict__ W,     // [N, 256] row-major
                    const float* __restrict__ bias,  // [N]
                    const float* __restrict__ zeta,  // [32] or nullptr
                    float* __restrict__ C,           // [M, N]
                    int M, int N)
{
    __shared__ float As[32 * KDIM];  // 32 KB: two 16-row A tiles

    const int tid  = threadIdx.x;
    const int wave = tid >> 5;
    const int lane = tid & 31;
    const int m0   = blockIdx.y * 32;
    const int n0   = blockIdx.x * 128 + wave * 16;

    // Cooperatively stage the 32 x 256 A tile (2048 float4 by 256 threads).
    #pragma unroll
    for (int it = 0; it < 8; ++it) {
        int i  = tid + it * 256;       // 0..2047
        int r  = i >> 6;               // row in tile (64 float4 per row)
        int c4 = i & 63;
        int row = m0 + r;
        if (row >= M) row = M - 1;     // clamp: duplicate row, discarded at store
        ((float4*)As)[i] = ((const float4*)(A + (size_t)row * KDIM))[c4];
    }
    __syncthreads();

    const int rl = lane & 15;          // M (for A) / N (for B) within tile
    const int kb = (lane >> 4) * 2;    // K pair base for this half-wave

    const float* arow0 = As + rl * KDIM + kb;              // M-tile 0
    const float* arow1 = As + (16 + rl) * KDIM + kb;       // M-tile 1
    const float* wrow  = W + (size_t)(n0 + rl) * KDIM + kb;

    v8f acc0 = {};
    v8f acc1 = {};
    #pragma unroll 8
    for (int k = 0; k < KDIM; k += 4) {
        v2f b  = *(const v2f*)(wrow + k);    // global_load_b64 (shared by 2 wmma)
        v2f a0 = *(const v2f*)(arow0 + k);   // ds_load_b64
        v2f a1 = *(const v2f*)(arow1 + k);   // ds_load_b64
        acc0 = __builtin_amdgcn_wmma_f32_16x16x4_f32(
            false, a0, false, b, (short)0, acc0, false, false);
        acc1 = __builtin_amdgcn_wmma_f32_16x16x4_f32(
            false, a1, false, b, (short)0, acc1, false, false);
    }

    // Epilogue: bias (+ optional zeta gate), guarded strided stores.
    const int n  = n0 + rl;
    const float bb = bias[n];
    const float zz = zeta ? zeta[n & (HEAD_DIM - 1)] : 1.0f;
    const int mbase0 = m0 + (lane >> 4) * 8;
    const int mbase1 = mbase0 + 16;
    #pragma unroll
    for (int r = 0; r < 8; ++r) {
        int m = mbase0 + r;
        if (m < M) C[(size_t)m * N + n] = (acc0[r] + bb) * zz;
    }
    #pragma unroll
    for (int r = 0; r < 8; ++r) {
        int m = mbase1 + r;
        if (m < M) C[(size_t)m * N + n] = (acc1[r] + bb) * zz;
    }
}

// ---------------------------------------------------------------------------
// Bilinear sampling + softmax + weighted accumulation.
// grid.x = BS*LEN_Q (one block per query), block = 256 (8 waves = 8 heads).
// Lane = channel. grid_sample(align_corners=False) simplifies to:
//   x = ref_x*W + off_x - 0.5 ,  y = ref_y*H + off_y - 0.5
// since the offset normalizer equals (W,H) and grid = 2*loc - 1.
// ---------------------------------------------------------------------------
__global__ __launch_bounds__(256)
void ms_deform_sample(const float* __restrict__ vproj,    // [BS, LEN_Q, 8, 32]
                      const float* __restrict__ offraw,   // [BS*LEN_Q, 256]
                      const float* __restrict__ attnraw,  // [BS*LEN_Q, 128]
                      const float* __restrict__ refpts,   // [BS, LEN_Q, 4, 2]
                      float* __restrict__ outheads)       // [BS*LEN_Q, 256]
{
    const int bq = blockIdx.x;
    const int h  = threadIdx.x >> 5;
    const int c  = threadIdx.x & 31;
    const int b  = bq / LEN_Q;

    const int HH[N_LEVELS] = {100, 50, 25, 13};
    const int ST[N_LEVELS] = {0, 10000, 12500, 13125};

    // Softmax over the 16 (level,point) logits of this head (uniform per wave).
    const float* ar = attnraw + (size_t)bq * (N_HEAD * 16) + h * 16;
    float lg[16];
    float mx = -3.0e38f;
    #pragma unroll
    for (int i = 0; i < 16; ++i) { lg[i] = ar[i]; mx = fmaxf(mx, lg[i]); }
    float ssum = 0.0f;
    #pragma unroll
    for (int i = 0; i < 16; ++i) { lg[i] = __expf(lg[i] - mx); ssum += lg[i]; }
    const float inv = 1.0f / ssum;

    const float* orow = offraw + (size_t)bq * (N_HEAD * 32) + h * 32;  // [4][4][2]
    const float* rp   = refpts + (size_t)bq * (N_LEVELS * 2);          // [4][2]
    const float* vb   = vproj + (size_t)b * LEN_Q * EMBED_DIM + h * HEAD_DIM + c;

    float acc = 0.0f;
    #pragma unroll
    for (int lvl = 0; lvl < N_LEVELS; ++lvl) {
        const int   wI = HH[lvl], hI = HH[lvl];   // square levels
        const float Wf = (float)wI, Hf = (float)hI;
        const float rx = rp[lvl * 2 + 0];
        const float ry = rp[lvl * 2 + 1];
        const int   st = ST[lvl];
        #pragma unroll
        for (int p = 0; p < N_POINTS; ++p) {
            const float ox = orow[(lvl * N_POINTS + p) * 2 + 0];
            const float oy = orow[(lvl * N_POINTS + p) * 2 + 1];
            const float x = rx * Wf + ox - 0.5f;
            const float y = ry * Hf + oy - 0.5f;
            const float xf = floorf(x), yf = floorf(y);
            const float wx = x - xf,    wy = y - yf;
            const int x0 = (int)xf, y0 = (int)yf;
            const int x1 = x0 + 1,  y1 = y0 + 1;
            const float aw = lg[lvl * N_POINTS + p] * inv;

            const bool bx0 = (x0 >= 0) & (x0 < wI);
            const bool bx1 = (x1 >= 0) & (x1 < wI);
            const bool by0 = (y0 >= 0) & (y0 < hI);
            const bool by1 = (y1 >= 0) & (y1 < hI);

            float v00 = 0.f, v01 = 0.f, v10 = 0.f, v11 = 0.f;
            if (bx0 & by0) v00 = vb[(size_t)(st + y0 * wI + x0) * EMBED_DIM];
            if (bx1 & by0) v01 = vb[(size_t)(st + y0 * wI + x1) * EMBED_DIM];
            if (bx0 & by1) v10 = vb[(size_t)(st + y1 * wI + x0) * EMBED_DIM];
            if (bx1 & by1) v11 = vb[(size_t)(st + y1 * wI + x1) * EMBED_DIM];

            const float top = v00 * (1.0f - wx) + v01 * wx;
            const float bot = v10 * (1.0f - wx) + v11 * wx;
            acc += aw * (top * (1.0f - wy) + bot * wy);
        }
    }
    outheads[(size_t)bq * EMBED_DIM + h * HEAD_DIM + c] = acc;
}

// ---------------------------------------------------------------------------
extern "C" void kernel_launch(void* const* d_in, const int* in_sizes, int n_in,
                              void* d_out, int out_size, void* d_ws, size_t ws_size,
                              hipStream_t stream)
{
    const float* query   = (const float*)d_in[0];
    const float* refpts  = (const float*)d_in[1];
    const float* value   = (const float*)d_in[2];
    // d_in[3] = value_spatial_shapes (static, hardcoded)
    const float* W_value = (const float*)d_in[4];
    const float* b_value = (const float*)d_in[5];
    const float* W_off   = (const float*)d_in[6];
    const float* b_off   = (const float*)d_in[7];
    const float* W_attn  = (const float*)d_in[8];
    const float* b_attn  = (const float*)d_in[9];
    const float* W_out   = (const float*)d_in[10];
    const float* b_out   = (const float*)d_in[11];
    const float* zeta    = (const float*)d_in[12];
    float* out = (float*)d_out;

    const int M = BS * LEN_Q;             // 26588
    const int Mtiles32 = (M + 31) / 32;   // 831

    // Workspace layout (fp32):
    float* vproj    = (float*)d_ws;                       // [M, 256]
    float* offraw   = vproj  + (size_t)M * 256;           // [M, 256]
    float* attnraw  = offraw + (size_t)M * 256;           // [M, 128]
    float* outheads = attnraw + (size_t)M * 128;          // [M, 256]

    dim3 blk(256);

    // 1) value projection (+ zeta gate fused)
    wmma_gemm_bias<<<dim3(EMBED_DIM / 128, Mtiles32), blk, 0, stream>>>(
        value, W_value, b_value, zeta, vproj, M, EMBED_DIM);

    // 2) sampling offsets
    wmma_gemm_bias<<<dim3(EMBED_DIM / 128, Mtiles32), blk, 0, stream>>>(
        query, W_off, b_off, nullptr, offraw, M, EMBED_DIM);

    // 3) attention logits (N = 128)
    wmma_gemm_bias<<<dim3(1, Mtiles32), blk, 0, stream>>>(
        query, W_attn, b_attn, nullptr, attnraw, M, 128);

    // 4) softmax + bilinear sampling + per-head accumulate
    ms_deform_sample<<<dim3(M), blk, 0, stream>>>(
        vproj, offraw, attnraw, refpts, outheads);

    // 5) output projection
    wmma_gemm_bias<<<dim3(EMBED_DIM / 128, Mtiles32), blk, 0, stream>>>(
        outheads, W_out, b_out, nullptr, out, M, EMBED_DIM);
}